// pose3d_online_2_45707041964538
// MI455X (gfx1250) — compile-verified
//
#include <hip/hip_runtime.h>
#include <hip/hip_bf16.h>
#include <math.h>

typedef __attribute__((ext_vector_type(2))) float v2f;
typedef __attribute__((ext_vector_type(8))) float v8f;

// MPI skeleton bones (14), BONES[:,0] and BONES[:,1]
__constant__ int kBoneA[14] = {0,1,2,3,1,5,6,1,14, 8, 9,14,11,12};
__constant__ int kBoneB[14] = {1,2,3,4,5,6,7,14,8, 9,10,11,12,13};

__device__ __forceinline__ float wave_sum32(float v) {
    v += __shfl_xor(v, 16, 32);
    v += __shfl_xor(v,  8, 32);
    v += __shfl_xor(v,  4, 32);
    v += __shfl_xor(v,  2, 32);
    v += __shfl_xor(v,  1, 32);
    return v;
}

__global__ __launch_bounds__(32)
void pose3d_losses_kernel(const float* __restrict__ pose3d,    // [6,3,15]
                          const float* __restrict__ bone_len,  // [14]
                          const float* __restrict__ pose2d,    // [2,15]
                          const float* __restrict__ R,         // [3,3] row-major
                          const float* __restrict__ C,         // [3,1]
                          const float* __restrict__ lift_dirs, // [3,14]
                          const int*   __restrict__ qptr,      // scalar
                          float* __restrict__ out)             // [4]
{
    const int lane = threadIdx.x;   // wave32: 0..31
    const int half = lane >> 4;     // K-half selector for WMMA operands
    const int lm   = lane & 15;     // M (A) / N (B,D) index

    int q = qptr[0];
    q = q < 0 ? 0 : (q > 5 ? 5 : q);
    const float* p = pose3d + q * 45;     // current pose [3,15]

    // ================= loss_proj: cam = R^T @ (p - C) via V_WMMA_F32_16X16X4_F32
    // A[m][k] = R^T[m][k] = R[k*3+m]  (16x4, M=0..2 used, K=0..2 used)
    // B[k][n] = (p - C)[k][n]         (4x16, N=0..14 used)
    // K placement: lanes 0-15 hold K=0 (vec.x) and K=1 (vec.y);
    //              lanes 16-31 hold K=2 (vec.x) and K=3 (vec.y, zero).
    float a0 = 0.f, a1 = 0.f, b0 = 0.f, b1 = 0.f;
    if (half == 0) {
        if (lm < 3)  { a0 = R[0*3 + lm];            a1 = R[1*3 + lm]; }
        if (lm < 15) { b0 = p[0*15 + lm] - C[0];    b1 = p[1*15 + lm] - C[1]; }
    } else {
        if (lm < 3)  { a0 = R[2*3 + lm]; }
        if (lm < 15) { b0 = p[2*15 + lm] - C[2]; }
    }
    v2f A; A.x = a0; A.y = a1;
    v2f B; B.x = b0; B.y = b1;

    float camx, camy, camz;
#if __has_builtin(__builtin_amdgcn_wmma_f32_16x16x4_f32)
    {
        v8f acc = {};
        // 8 args: (neg_a, A, neg_b, B, c_mod, C, reuse_a, reuse_b); EXEC is all-1s here.
        acc = __builtin_amdgcn_wmma_f32_16x16x4_f32(false, A, false, B,
                                                    (short)0, acc, false, false);
        // D layout: VGPR r -> M=r, lanes 0-15 -> N=lane  => rows 0..2 = cam x/y/z
        camx = acc[0]; camy = acc[1]; camz = acc[2];
    }
#else
    {
        camx = 0.f; camy = 0.f; camz = 1.f;
        if (half == 0 && lm < 15) {
            float x0 = p[lm] - C[0], x1 = p[15 + lm] - C[1], x2 = p[30 + lm] - C[2];
            camx = R[0]*x0 + R[3]*x1 + R[6]*x2;
            camy = R[1]*x0 + R[4]*x1 + R[7]*x2;
            camz = R[2]*x0 + R[5]*x1 + R[8]*x2;
        }
    }
#endif

    float lp = 0.f;
    if (half == 0 && lm < 15) {
        float ix = 512.0f * camx / camz + 512.0f;
        float iy = 512.0f * camy / camz + 288.0f;
        float dx = ix - pose2d[lm];
        float dy = iy - pose2d[15 + lm];
        lp = dx*dx + dy*dy;
    }
    lp = wave_sum32(lp) * (1.0f / 30.0f);

    // ================= loss_smooth (forward term iff q!=5, backward iff q!=0)
    float ssum = 0.f;
    if (q != 5) {
        const float* pa = pose3d + q * 45;
        const float* pb = pose3d + (q + 1) * 45;
        float d0 = pa[lane] - pb[lane];
        ssum += d0 * d0;
        if (lane + 32 < 45) { float d1 = pa[lane+32] - pb[lane+32]; ssum += d1*d1; }
    }
    if (q != 0) {
        const float* pa = pose3d + (q - 1) * 45;
        const float* pb = pose3d + q * 45;
        float d0 = pa[lane] - pb[lane];
        ssum += d0 * d0;
        if (lane + 32 < 45) { float d1 = pa[lane+32] - pb[lane+32]; ssum += d1*d1; }
    }
    float ls = wave_sum32(ssum) * (1.0f / 45.0f);

    // ================= loss_bone + loss_lift (one bone per lane, 14 bones)
    float lb = 0.f, ll = 0.f;
    if (lane < 14) {
        int j0 = kBoneA[lane], j1 = kBoneB[lane];
        float vx = p[j0]      - p[j1];
        float vy = p[15 + j0] - p[15 + j1];
        float vz = p[30 + j0] - p[30 + j1];
        float sq = vx*vx + vy*vy + vz*vz;
        float e  = bone_len[lane] - sq;        // reference uses SQUARED length
        lb = e * e;
        float inv = 1.0f / (sqrtf(sq) + 1e-5f);
        float dx = vx*inv - lift_dirs[lane];
        float dy = vy*inv - lift_dirs[14 + lane];
        float dz = vz*inv - lift_dirs[28 + lane];
        ll = dx*dx + dy*dy + dz*dz;
    }
    lb = wave_sum32(lb) * (1.0f / 14.0f);
    ll = wave_sum32(ll) * (1.0f / 42.0f);

    if (lane == 0) {
        out[0] = lp;
        out[1] = ls;
        out[2] = lb;
        out[3] = ll;
    }
}

extern "C" void kernel_launch(void* const* d_in, const int* in_sizes, int n_in,
                              void* d_out, int out_size, void* d_ws, size_t ws_size,
                              hipStream_t stream) {
    (void)in_sizes; (void)n_in; (void)out_size; (void)d_ws; (void)ws_size;
    pose3d_losses_kernel<<<1, 32, 0, stream>>>(
        (const float*)d_in[0],   // pose3d [6,3,15]
        (const float*)d_in[1],   // bone_lengths [14]
        (const float*)d_in[2],   // pose_2d [2,15]
        (const float*)d_in[3],   // R_drone [3,3]
        (const float*)d_in[4],   // C_drone [3,1]
        (const float*)d_in[5],   // lift directions [3,14]
        (const int*)d_in[6],     // queue_index
        (float*)d_out);          // [4] losses
}